// SupConLoss_23287312679528
// MI455X (gfx1250) — compile-verified
//
#include <hip/hip_runtime.h>
#include <hip/hip_bf16.h>
#include <math.h>

typedef __attribute__((ext_vector_type(2))) float v2f;
typedef __attribute__((ext_vector_type(4))) float v4f;
typedef __attribute__((ext_vector_type(8))) float v8f;
typedef __attribute__((ext_vector_type(4))) int   v4i;

typedef __attribute__((address_space(1))) v4i gas_v4i;  // global
typedef __attribute__((address_space(3))) v4i las_v4i;  // LDS

#define NSAMP 4096          // B
#define NROWS 8192          // V*B
#define DIMS  128
#define CHUNK 32            // columns staged per LDS buffer
#define INV_T (1.0f / 0.07f)

#if __has_builtin(__builtin_amdgcn_global_load_async_to_lds_b128)
#define USE_ASYNC_LDS 1
#else
#define USE_ASYNC_LDS 0
#endif

// features layout: (B=4096, V=2, D=128) row-major.
// contrast row i (view-major) -> features[(i & 4095)][i >> 12][:]
__device__ __forceinline__ const float* contrast_row(const float* feat, int i) {
    return feat + (size_t)(i & (NSAMP - 1)) * (2 * DIMS) + (size_t)(i >> 12) * DIMS;
}

// Stage CHUNK contrast rows (CHUNK x 128 f32) into LDS. Each of 256 threads
// moves 4 x 16B. Async path: direct global->LDS DMA tracked by ASYNCcnt.
__device__ __forceinline__ void stage_chunk(const float* feat, float* dst,
                                            int chunk, int tid) {
#pragma unroll
    for (int it = 0; it < 4; ++it) {
        int idx = tid + 256 * it;
        int row = idx >> 5;      // 0..CHUNK-1
        int f4  = idx & 31;      // float4 index within a 128-f32 row
        const float* src = contrast_row(feat, chunk + row) + f4 * 4;
        float* d = dst + row * DIMS + f4 * 4;
#if USE_ASYNC_LDS
        __builtin_amdgcn_global_load_async_to_lds_b128(
            (gas_v4i*)src, (las_v4i*)d, 0, 0);
#else
        *(v4f*)d = *(const v4f*)src;
#endif
    }
}

__device__ __forceinline__ void wait_async0() {
#if USE_ASYNC_LDS
#if __has_builtin(__builtin_amdgcn_s_wait_asynccnt)
    __builtin_amdgcn_s_wait_asynccnt(0);
#else
    asm volatile("s_wait_asynccnt 0x0" ::: "memory");
#endif
#endif
}

// 16x16 f32 logits tile: C += A(16x128) * B(128x16) via 32 WMMA k-steps.
// B fragments are preloaded in two 16-step bursts so DS loads issue ahead
// of the WMMA chain (one dscnt wait per burst instead of per pair).
__device__ __forceinline__ v8f tile_mm(const v2f* areg, const float* brow, int hi) {
    v8f c = {};
#pragma unroll
    for (int half = 0; half < 2; ++half) {
        v2f breg[16];
#pragma unroll
        for (int k = 0; k < 16; ++k)
            breg[k] = *(const v2f*)(brow + 4 * (16 * half + k) + 2 * hi);
#pragma unroll
        for (int k = 0; k < 16; ++k)
            c = __builtin_amdgcn_wmma_f32_16x16x4_f32(
                    false, areg[16 * half + k], false, breg[k],
                    (short)0, c, false, false);
    }
    return c;
}

__global__ void supcon_pack_targets(const int* __restrict__ targets,
                                    unsigned* __restrict__ packed) {
    int b = blockIdx.x * 256 + threadIdx.x;
    if (b < NSAMP) {
        unsigned m = 0;
#pragma unroll
        for (int k = 0; k < 6; ++k)
            m |= ((unsigned)targets[b * 6 + k] & 1u) << k;
        packed[b] = m;
    }
}

__global__ void supcon_zero_out(float* out) {
    if (threadIdx.x == 0 && blockIdx.x == 0) out[0] = 0.0f;
}

__global__ __launch_bounds__(256)
void supcon_main(const float* __restrict__ feat,
                 const unsigned* __restrict__ packed,
                 float* __restrict__ out) {
    __shared__ float    ldsB[2][CHUNK * DIMS];  // double-buffered column chunks
    __shared__ unsigned ldsP[NSAMP];            // packed targets, 16 KB

    const int tid  = threadIdx.x;
    const int wave = tid >> 5;     // 0..7
    const int lane = tid & 31;
    const int hi   = lane >> 4;    // lane half (0/1)
    const int l16  = lane & 15;

    // stage packed targets into LDS (plain stores; covered by first barrier)
    for (int i = tid; i < NSAMP; i += 256) ldsP[i] = packed[i];

    const int rowBase = blockIdx.x * 128 + wave * 16;

    // ---- Preload A tile (16 rows x 128) into registers, WMMA 16x4 f32 A layout:
    // lane L: M = L%16; per k-step VGPR pair holds K = 4k + 2*(L/16) + {0,1}
    const float* arow = contrast_row(feat, rowBase + l16);
    v2f areg[32];
#pragma unroll
    for (int k = 0; k < 32; ++k)
        areg[k] = *(const v2f*)(arow + 4 * k + 2 * hi);

    // C layout: lane L holds rows (r + 8*(L/16)), cols = L%16 (per 16-col tile)
    float rowMax[8];
#pragma unroll
    for (int r = 0; r < 8; ++r) rowMax[r] = -3.0e38f;

    // =================== Pass A: row maxima of logits ===================
    stage_chunk(feat, ldsB[0], 0, tid);
    for (int chunk = 0; chunk < NROWS; chunk += CHUNK) {
        const int sel = (chunk / CHUNK) & 1;
        wait_async0();
        __syncthreads();                       // buffer[sel] ready everywhere
        if (chunk + CHUNK < NROWS)             // prefetch next chunk (overlap)
            stage_chunk(feat, ldsB[sel ^ 1], chunk + CHUNK, tid);

#pragma unroll
        for (int sub = 0; sub < CHUNK / 16; ++sub) {
            const float* brow = &ldsB[sel][(sub * 16 + l16) * DIMS];
            v8f c = tile_mm(areg, brow, hi);
#pragma unroll
            for (int r = 0; r < 8; ++r)
                rowMax[r] = fmaxf(rowMax[r], c[r] * INV_T);
        }
    }
    // reduce max across the 16-lane group holding each row
#pragma unroll
    for (int r = 0; r < 8; ++r) {
        float m = rowMax[r];
#pragma unroll
        for (int off = 8; off >= 1; off >>= 1)
            m = fmaxf(m, __shfl_xor(m, off, 16));
        rowMax[r] = m;
    }

    // =================== Pass B: weighted exp-sums + positive logit =====
    float sumExp[8], posL[8];
    unsigned pi[8];
#pragma unroll
    for (int r = 0; r < 8; ++r) {
        sumExp[r] = 0.0f;
        posL[r]   = 0.0f;
        int i = rowBase + r + 8 * hi;
        pi[r] = ldsP[i & (NSAMP - 1)];
    }

    stage_chunk(feat, ldsB[0], 0, tid);
    for (int chunk = 0; chunk < NROWS; chunk += CHUNK) {
        const int sel = (chunk / CHUNK) & 1;
        wait_async0();
        __syncthreads();
        if (chunk + CHUNK < NROWS)
            stage_chunk(feat, ldsB[sel ^ 1], chunk + CHUNK, tid);

#pragma unroll
        for (int sub = 0; sub < CHUNK / 16; ++sub) {
            const int j = chunk + sub * 16 + l16;     // this lane's column
            const unsigned pj = ldsP[j & (NSAMP - 1)];
            const float* brow = &ldsB[sel][(sub * 16 + l16) * DIMS];
            v8f c = tile_mm(areg, brow, hi);
#pragma unroll
            for (int r = 0; r < 8; ++r) {
                int   i = rowBase + r + 8 * hi;
                float l = c[r] * INV_T;
                if (j == (i ^ NSAMP)) posL[r] += l;    // the single positive
                float w;
                if (j == i)                                      w = 0.0f;
                else if ((j & (NSAMP - 1)) == (i & (NSAMP - 1))) w = 1.0f;
                else w = (float)__popc(pi[r] ^ pj);    // 0 if identical targets
                sumExp[r] += w * __expf(l - rowMax[r]);
            }
        }
    }

    // reduce sums / positive logits across 16-lane groups
#pragma unroll
    for (int r = 0; r < 8; ++r) {
        float s = sumExp[r], p = posL[r];
#pragma unroll
        for (int off = 8; off >= 1; off >>= 1) {
            s += __shfl_xor(s, off, 16);
            p += __shfl_xor(p, off, 16);
        }
        sumExp[r] = s;
        posL[r]   = p;
    }

    if (l16 == 0) {
        float acc = 0.0f;
#pragma unroll
        for (int r = 0; r < 8; ++r)
            acc += -(posL[r] - rowMax[r] - __logf(sumExp[r]));
        atomicAdd(out, acc * (1.0f / (float)NROWS));
    }
}

extern "C" void kernel_launch(void* const* d_in, const int* in_sizes, int n_in,
                              void* d_out, int out_size, void* d_ws, size_t ws_size,
                              hipStream_t stream) {
    const float* feat    = (const float*)d_in[0];   // (4096, 2, 128) f32
    const int*   targets = (const int*)d_in[1];     // (4096, 6) i32
    float*       out     = (float*)d_out;           // scalar
    unsigned*    packed  = (unsigned*)d_ws;         // 4096 u32 scratch

    supcon_pack_targets<<<(NSAMP + 255) / 256, 256, 0, stream>>>(targets, packed);
    supcon_zero_out<<<1, 32, 0, stream>>>(out);
    supcon_main<<<NROWS / 128, 256, 0, stream>>>(feat, packed, out);
}